// GAT_50697793962251
// MI455X (gfx1250) — compile-verified
//
#include <hip/hip_runtime.h>

typedef float v2f __attribute__((ext_vector_type(2)));
typedef float v8f __attribute__((ext_vector_type(8)));

constexpr int E_NODES = 131072;          // line-graph nodes
constexpr int M_EDGES = 2097152;         // line-graph edges
constexpr int MTOT    = M_EDGES + E_NODES; // + self loops
constexpr int NNODES  = 16384;
constexpr int IN_F    = 64;
constexpr int HC      = 256;             // HEADS*CH
constexpr int CH      = 64;

// ---- order-preserving float<->uint map for atomicMax on floats ----
__device__ __forceinline__ unsigned fenc(float f) {
  unsigned u = __float_as_uint(f);
  return (u & 0x80000000u) ? ~u : (u | 0x80000000u);
}
__device__ __forceinline__ float fdec(unsigned u) {
  u = (u & 0x80000000u) ? (u & 0x7fffffffu) : ~u;
  return __uint_as_float(u);
}
__device__ __forceinline__ float lrelu(float v) { return v >= 0.f ? v : 0.2f * v; }

__device__ __forceinline__ void edge_sd(const int* __restrict__ ei, int m, int& src, int& dst) {
  if (m < M_EDGES) { src = ei[m]; dst = ei[M_EDGES + m]; }
  else             { src = m - M_EDGES; dst = src; }
}

// ================= Kernel 1: h = x @ W^T  (E x 64) x (64 x 256) =================
// fp32 WMMA 16x16x4. Block = 16 rows x 256 cols; 8 waves, each wave does 2 col tiles.
__global__ void __launch_bounds__(256)
k_gemm_h(const float* __restrict__ x, const float* __restrict__ W, float* __restrict__ h)
{
  __shared__ float sWt[IN_F * HC];       // sWt[k*256 + o] = W[o*64 + k]  (64 KB)
  const int tid = threadIdx.x;
  for (int j = tid; j < IN_F * HC; j += 256) {
    const int k = j >> 8, o = j & 255;   // destination-major: conflict-free LDS stores
    sWt[j] = W[o * IN_F + k];
  }
  __syncthreads();

  const int lane    = tid & 31;
  const int wv      = tid >> 5;
  const int rowBase = blockIdx.x * 16;
  const int mlane   = lane & 15;
  const int khalf   = (lane >> 4) ? 2 : 0;  // lanes 16-31 carry K = k+2, k+3
  const float* xr   = x + (size_t)(rowBase + mlane) * IN_F;

  #pragma unroll
  for (int t = 0; t < 2; ++t) {
    const int colBase = (wv + t * 8) * 16;
    v8f c = {};
    #pragma unroll
    for (int s = 0; s < 16; ++s) {
      const int koff = s * 4 + khalf;
      v2f a = *(const v2f*)(xr + koff);                       // A: x[row][koff..koff+1]
      v2f b;
      b[0] = sWt[koff * HC + colBase + mlane];                // B: W^T[koff][col]
      b[1] = sWt[(koff + 1) * HC + colBase + mlane];
      c = __builtin_amdgcn_wmma_f32_16x16x4_f32(false, a, false, b, (short)0, c, false, false);
    }
    const int mrow0 = (lane >> 4) ? 8 : 0;
    #pragma unroll
    for (int r = 0; r < 8; ++r)
      h[(size_t)(rowBase + mrow0 + r) * HC + colBase + mlane] = c[r];
  }
}

// ================= Kernel 2: a_src/a_dst per (e, head) =================
// One wave per edge-node row: 32 lanes x 8 f32 = 256 values; head = lane>>3.
__global__ void __launch_bounds__(256)
k_attdot(const float* __restrict__ h, const float* __restrict__ att_src,
         const float* __restrict__ att_dst, float* __restrict__ a_src, float* __restrict__ a_dst)
{
  const int lane = threadIdx.x & 31;
  const int wv   = threadIdx.x >> 5;
  const int e    = blockIdx.x * 8 + wv;
  const size_t base = (size_t)e * HC + lane * 8;
  const float4 u0 = *(const float4*)(h + base);
  const float4 u1 = *(const float4*)(h + base + 4);
  const int aoff  = lane * 8;    // == head*64 + (lane&7)*8
  const float4 s0 = *(const float4*)(att_src + aoff);
  const float4 s1 = *(const float4*)(att_src + aoff + 4);
  const float4 d0 = *(const float4*)(att_dst + aoff);
  const float4 d1 = *(const float4*)(att_dst + aoff + 4);
  float ps = u0.x*s0.x + u0.y*s0.y + u0.z*s0.z + u0.w*s0.w
           + u1.x*s1.x + u1.y*s1.y + u1.z*s1.z + u1.w*s1.w;
  float pd = u0.x*d0.x + u0.y*d0.y + u0.z*d0.z + u0.w*d0.w
           + u1.x*d1.x + u1.y*d1.y + u1.z*d1.z + u1.w*d1.w;
  #pragma unroll
  for (int msk = 1; msk < 8; msk <<= 1) {
    ps += __shfl_xor(ps, msk, 32);
    pd += __shfl_xor(pd, msk, 32);
  }
  if ((lane & 7) == 0) {
    const int head = lane >> 3;
    a_src[e * 4 + head] = ps;
    a_dst[e * 4 + head] = pd;
  }
}

// ================= Kernel 3: segment max of logits per dst =================
__global__ void __launch_bounds__(256)
k_edge_max(const int* __restrict__ ei, const float* __restrict__ a_src,
           const float* __restrict__ a_dst, unsigned* __restrict__ lmax)
{
  const int m = blockIdx.x * 256 + threadIdx.x;
  if (m >= MTOT) return;
  int src, dst; edge_sd(ei, m, src, dst);
  const float4 as = *(const float4*)(a_src + src * 4);
  const float4 ad = *(const float4*)(a_dst + dst * 4);
  atomicMax(lmax + dst * 4 + 0, fenc(lrelu(as.x + ad.x)));
  atomicMax(lmax + dst * 4 + 1, fenc(lrelu(as.y + ad.y)));
  atomicMax(lmax + dst * 4 + 2, fenc(lrelu(as.z + ad.z)));
  atomicMax(lmax + dst * 4 + 3, fenc(lrelu(as.w + ad.w)));
}

// ================= Kernel 4: segment sum of exp(logit - max) =================
__global__ void __launch_bounds__(256)
k_edge_sum(const int* __restrict__ ei, const float* __restrict__ a_src,
           const float* __restrict__ a_dst, const unsigned* __restrict__ lmax,
           float* __restrict__ denom)
{
  const int m = blockIdx.x * 256 + threadIdx.x;
  if (m >= MTOT) return;
  int src, dst; edge_sd(ei, m, src, dst);
  const float4 as = *(const float4*)(a_src + src * 4);
  const float4 ad = *(const float4*)(a_dst + dst * 4);
  const unsigned* lm = lmax + dst * 4;
  atomicAdd(denom + dst * 4 + 0, __expf(lrelu(as.x + ad.x) - fdec(lm[0])));
  atomicAdd(denom + dst * 4 + 1, __expf(lrelu(as.y + ad.y) - fdec(lm[1])));
  atomicAdd(denom + dst * 4 + 2, __expf(lrelu(as.z + ad.z) - fdec(lm[2])));
  atomicAdd(denom + dst * 4 + 3, __expf(lrelu(as.w + ad.w) - fdec(lm[3])));
}

// ======= Kernel 5: weighted scatter, head-mean folded in (4x fewer atomics) =======
// One wave per edge: gather h[src] (1 KB, coalesced per head), accumulate
// 0.25 * sum_h alpha_h * h[src][h][c] into outm[dst][c].
__global__ void __launch_bounds__(256)
k_scatter(const int* __restrict__ ei, const float* __restrict__ a_src,
          const float* __restrict__ a_dst, const unsigned* __restrict__ lmax,
          const float* __restrict__ denom, const float* __restrict__ h,
          float* __restrict__ outm)
{
  const int lane = threadIdx.x & 31;
  const int wv   = threadIdx.x >> 5;
  const int m    = blockIdx.x * 8 + wv;
  int src, dst; edge_sd(ei, m, src, dst);
  const float4 as = *(const float4*)(a_src + src * 4);
  const float4 ad = *(const float4*)(a_dst + dst * 4);
  const unsigned* lm = lmax + dst * 4;
  const float4 dn = *(const float4*)(denom + dst * 4);
  const float al0 = __expf(lrelu(as.x + ad.x) - fdec(lm[0])) / (dn.x + 1e-16f);
  const float al1 = __expf(lrelu(as.y + ad.y) - fdec(lm[1])) / (dn.y + 1e-16f);
  const float al2 = __expf(lrelu(as.z + ad.z) - fdec(lm[2])) / (dn.z + 1e-16f);
  const float al3 = __expf(lrelu(as.w + ad.w) - fdec(lm[3])) / (dn.w + 1e-16f);
  const float* hs = h + (size_t)src * HC;
  #pragma unroll
  for (int t = 0; t < 2; ++t) {
    const int c = lane + t * 32;
    const float v = 0.25f * (al0 * hs[c] + al1 * hs[64 + c] + al2 * hs[128 + c] + al3 * hs[192 + c]);
    atomicAdd(outm + (size_t)dst * CH + c, v);
  }
}

// ================= Kernel 6: scatter-mean onto graph nodes =================
__global__ void __launch_bounds__(256)
k_nodes(const float* __restrict__ outm, const float* __restrict__ bias,
        const int* __restrict__ rows, float* __restrict__ agg, float* __restrict__ deg)
{
  const int lane = threadIdx.x & 31;
  const int wv   = threadIdx.x >> 5;
  const int e    = blockIdx.x * 8 + wv;
  const int r    = rows[e];
  #pragma unroll
  for (int t = 0; t < 2; ++t) {
    const int c = lane + t * 32;
    atomicAdd(agg + (size_t)r * CH + c, outm[(size_t)e * CH + c] + bias[c]);
  }
  if (lane == 0) atomicAdd(deg + r, 1.0f);
}

__global__ void __launch_bounds__(256)
k_norm(float* __restrict__ agg, const float* __restrict__ deg)
{
  const int i = blockIdx.x * 256 + threadIdx.x;
  const float d = deg[i >> 6];
  agg[i] = (d > 0.f) ? agg[i] / d : 0.f;
}

// ================= Kernel 7: out = agg @ w_weight^T + w_bias =================
// (16384 x 64) x (64 x 64), fp32 WMMA 16x16x4. Block: 32 rows x 64 cols, 8 waves.
__global__ void __launch_bounds__(256)
k_gemm_out(const float* __restrict__ agg, const float* __restrict__ ww,
           const float* __restrict__ wb, float* __restrict__ out)
{
  __shared__ float sWt[CH * CH];          // sWt[k*64 + o] = ww[o*64 + k]  (16 KB)
  const int tid = threadIdx.x;
  for (int j = tid; j < CH * CH; j += 256) {
    const int k = j >> 6, o = j & 63;
    sWt[j] = ww[o * CH + k];
  }
  __syncthreads();

  const int lane    = tid & 31;
  const int wv      = tid >> 5;
  const int rowBase = blockIdx.x * 32 + (wv >> 2) * 16;
  const int colBase = (wv & 3) * 16;
  const int mlane   = lane & 15;
  const int khalf   = (lane >> 4) ? 2 : 0;
  const float* ar   = agg + (size_t)(rowBase + mlane) * CH;

  v8f c = {};
  #pragma unroll
  for (int s = 0; s < 16; ++s) {
    const int koff = s * 4 + khalf;
    v2f a = *(const v2f*)(ar + koff);
    v2f b;
    b[0] = sWt[koff * CH + colBase + mlane];
    b[1] = sWt[(koff + 1) * CH + colBase + mlane];
    c = __builtin_amdgcn_wmma_f32_16x16x4_f32(false, a, false, b, (short)0, c, false, false);
  }
  const float bv   = wb[colBase + mlane];
  const int mrow0  = (lane >> 4) ? 8 : 0;
  #pragma unroll
  for (int r = 0; r < 8; ++r)
    out[(size_t)(rowBase + mrow0 + r) * CH + colBase + mlane] = c[r] + bv;
}

// =================================================================================
extern "C" void kernel_launch(void* const* d_in, const int* in_sizes, int n_in,
                              void* d_out, int out_size, void* d_ws, size_t ws_size,
                              hipStream_t stream)
{
  (void)in_sizes; (void)n_in; (void)out_size; (void)ws_size;
  const float* x       = (const float*)d_in[0];
  const int*   ei      = (const int*)  d_in[1];
  const int*   rows    = (const int*)  d_in[2];
  const float* W       = (const float*)d_in[3];
  const float* att_src = (const float*)d_in[4];
  const float* att_dst = (const float*)d_in[5];
  const float* bias    = (const float*)d_in[6];
  const float* ww      = (const float*)d_in[7];
  const float* wb      = (const float*)d_in[8];

  // workspace layout (floats); zero-init region first
  float*    ws    = (float*)d_ws;
  float*    outm  = ws;                                   // E*64
  float*    agg   = outm  + (size_t)E_NODES * CH;         // N*64
  float*    deg   = agg   + (size_t)NNODES  * CH;         // N
  float*    denom = deg   + NNODES;                       // E*4
  unsigned* lmax  = (unsigned*)(denom + (size_t)E_NODES * 4); // E*4 (0 is below enc(-inf), ok)
  float*    a_src = (float*)(lmax + (size_t)E_NODES * 4); // E*4
  float*    a_dst = a_src + (size_t)E_NODES * 4;          // E*4
  float*    h     = a_dst + (size_t)E_NODES * 4;          // E*256

  const size_t zero_bytes =
      ((size_t)E_NODES * CH + (size_t)NNODES * CH + NNODES +
       (size_t)E_NODES * 4 + (size_t)E_NODES * 4) * sizeof(float);
  hipMemsetAsync(d_ws, 0, zero_bytes, stream);

  k_gemm_h  <<<E_NODES / 16, 256, 0, stream>>>(x, W, h);
  k_attdot  <<<E_NODES / 8,  256, 0, stream>>>(h, att_src, att_dst, a_src, a_dst);
  k_edge_max<<<MTOT / 256,   256, 0, stream>>>(ei, a_src, a_dst, lmax);
  k_edge_sum<<<MTOT / 256,   256, 0, stream>>>(ei, a_src, a_dst, lmax, denom);
  k_scatter <<<MTOT / 8,     256, 0, stream>>>(ei, a_src, a_dst, lmax, denom, h, outm);
  k_nodes   <<<E_NODES / 8,  256, 0, stream>>>(outm, bias, rows, agg, deg);
  k_norm    <<<(NNODES * CH) / 256, 256, 0, stream>>>(agg, deg);
  k_gemm_out<<<NNODES / 32,  256, 0, stream>>>(agg, ww, wb, (float*)d_out);
}